// GCN_19258633355948
// MI455X (gfx1250) — compile-verified
//
#include <hip/hip_runtime.h>

typedef __attribute__((ext_vector_type(2))) float v2f;
typedef __attribute__((ext_vector_type(8))) float v8f;

// ---------------------------------------------------------------------------
// Degree / normalization kernels
// ---------------------------------------------------------------------------
__global__ __launch_bounds__(256)
void gcn_deg_init(float* __restrict__ deg, int n) {
    int i = blockIdx.x * blockDim.x + threadIdx.x;
    if (i < n) deg[i] = 1.0f;  // self-loop contributes 1 to every degree
}

__global__ __launch_bounds__(256)
void gcn_deg_count(const long long* __restrict__ dst, float* __restrict__ deg, int E) {
    int stride = gridDim.x * blockDim.x;
    for (int e = blockIdx.x * blockDim.x + threadIdx.x; e < E; e += stride)
        atomicAdd(&deg[(int)dst[e]], 1.0f);
}

__global__ __launch_bounds__(256)
void gcn_deg_rsqrt(const float* __restrict__ deg, float* __restrict__ dinv, int n) {
    int i = blockIdx.x * blockDim.x + threadIdx.x;
    if (i < n) dinv[i] = rsqrtf(deg[i]);
}

// ---------------------------------------------------------------------------
// Dense transform: H = (relu_in ? relu(A) : A) @ W   using fp32 WMMA 16x16x4.
// Fused epilogue: AGG = bias + dinv^2 * H   (self-loop message + bias init,
// so the scatter kernel only needs to add edge messages).
// One wave computes one 16(M) x 16(N) tile; 4 waves per block tile N.
// ---------------------------------------------------------------------------
__global__ __launch_bounds__(128)
void gcn_gemm_wmma(const float* __restrict__ A,     // [M, K]
                   const float* __restrict__ W,     // [K, N]
                   const float* __restrict__ bias,  // [N]
                   const float* __restrict__ dinv,  // [M]
                   float* __restrict__ H,           // [M, N]
                   float* __restrict__ AGG,         // [M, N]
                   int M, int K, int N, int relu_in)
{
    const int lane = threadIdx.x & 31;
    const int wave = threadIdx.x >> 5;
    const int half = lane >> 4;   // 0: lanes 0-15, 1: lanes 16-31
    const int lm   = lane & 15;

    const int mtile = blockIdx.x;
    const int ntile = blockIdx.y * 4 + wave;
    if (ntile * 16 >= N) return;

    const int row = min(mtile * 16 + lm, M - 1);  // A-fragment row for this lane
    const int col = ntile * 16 + lm;              // B/C column for this lane

    const float* __restrict__ arow = A + (size_t)row * K;

    v8f c = {};
    for (int k = 0; k < K; k += 4) {
        // 32-bit A 16x4 layout: VGPR0 = K={0,2}, VGPR1 = K={1,3} split by half-wave
        const int ka = k + 2 * half;
        float a0 = arow[ka];
        float a1 = arow[ka + 1];
        if (relu_in) { a0 = fmaxf(a0, 0.0f); a1 = fmaxf(a1, 0.0f); }
        v2f a; a.x = a0; a.y = a1;
        v2f b; b.x = W[(size_t)ka * N + col];
               b.y = W[(size_t)(ka + 1) * N + col];
        // D = A*B + C ; emits v_wmma_f32_16x16x4_f32
        c = __builtin_amdgcn_wmma_f32_16x16x4_f32(
                /*neg_a=*/false, a, /*neg_b=*/false, b,
                /*c_mod=*/(short)0, c, /*reuse_a=*/false, /*reuse_b=*/false);
    }

    const float bcol = bias[col];
#pragma unroll
    for (int r = 0; r < 8; ++r) {
        const int rr = mtile * 16 + 8 * half + r;   // C/D layout: VGPR r -> M = r + 8*half
        if (rr < M) {
            const size_t idx = (size_t)rr * N + col;
            const float v  = c[r];
            const float dv = dinv[rr];
            H[idx]   = v;
            AGG[idx] = bcol + dv * dv * v;          // self-loop + bias
        }
    }
}

// ---------------------------------------------------------------------------
// Edge scatter-aggregate: AGG[dst] += H[src] * (dinv[src]*dinv[dst]).
// One wave32 per edge; lanes cover the feature row with float4 gathers
// (coalesced, L2-resident) and scalar f32 global atomics.
// ---------------------------------------------------------------------------
__global__ __launch_bounds__(256)
void gcn_scatter(const long long* __restrict__ src,
                 const long long* __restrict__ dst,
                 const float* __restrict__ dinv,
                 const float* __restrict__ H,    // [M, F]
                 float* __restrict__ AGG,        // [M, F]
                 int E, int F4)                  // F4 = F/4
{
    const int gwave = (blockIdx.x * blockDim.x + threadIdx.x) >> 5;
    const int nwave = (gridDim.x * blockDim.x) >> 5;
    const int lane  = threadIdx.x & 31;
    const int F     = F4 * 4;

    for (int e = gwave; e < E; e += nwave) {
        const int s = (int)src[e];
        const int d = (int)dst[e];
        const float norm = dinv[s] * dinv[d];
        const float4* __restrict__ hrow = (const float4*)(H + (size_t)s * F);
        float* __restrict__ orow = AGG + (size_t)d * F;
        for (int j = lane; j < F4; j += 32) {
            const float4 v = hrow[j];
            atomicAdd(&orow[4 * j + 0], v.x * norm);
            atomicAdd(&orow[4 * j + 1], v.y * norm);
            atomicAdd(&orow[4 * j + 2], v.z * norm);
            atomicAdd(&orow[4 * j + 3], v.w * norm);
        }
    }
}

// ---------------------------------------------------------------------------
// Final head: out[i] = sum_f relu(AGG2[i][f]) * Wfc[f] + bfc.  One wave per
// node, 128 features = 4 per lane, wave32 shuffle reduction.
// ---------------------------------------------------------------------------
__global__ __launch_bounds__(256)
void gcn_head(const float* __restrict__ AGG2,   // [M, 128]
              const float* __restrict__ Wfc,    // [128]
              const float* __restrict__ bfc,    // [1]
              float* __restrict__ out, int M)
{
    const int gwave = (blockIdx.x * blockDim.x + threadIdx.x) >> 5;
    const int nwave = (gridDim.x * blockDim.x) >> 5;
    const int lane  = threadIdx.x & 31;

    for (int i = gwave; i < M; i += nwave) {
        const float* __restrict__ row = AGG2 + (size_t)i * 128;
        float s = 0.0f;
#pragma unroll
        for (int j = 0; j < 4; ++j) {
            const int f = lane + 32 * j;
            s += fmaxf(row[f], 0.0f) * Wfc[f];
        }
#pragma unroll
        for (int off = 16; off > 0; off >>= 1)
            s += __shfl_xor(s, off, 32);
        if (lane == 0) out[i] = s + bfc[0];
    }
}

// ---------------------------------------------------------------------------
// Launch
// ---------------------------------------------------------------------------
extern "C" void kernel_launch(void* const* d_in, const int* in_sizes, int n_in,
                              void* d_out, int out_size, void* d_ws, size_t ws_size,
                              hipStream_t stream)
{
    const float*     x   = (const float*)d_in[0];      // [n, 100]
    const long long* ei  = (const long long*)d_in[1];  // [2, E] int64
    const float*     W1  = (const float*)d_in[2];      // [100, 256]
    const float*     b1  = (const float*)d_in[3];      // [256]
    const float*     W2  = (const float*)d_in[4];      // [256, 128]
    const float*     b2  = (const float*)d_in[5];      // [128]
    const float*     Wfc = (const float*)d_in[6];      // [128, 1]
    const float*     bfc = (const float*)d_in[7];      // [1]
    float*           out = (float*)d_out;              // [n, 1]

    const int F_IN = 100, F1 = 256, F2 = 128;
    const int n = in_sizes[0] / F_IN;   // 50000
    const int E = in_sizes[1] / 2;      // 800000
    const long long* esrc = ei;
    const long long* edst = ei + E;

    // Workspace layout (all fp32)
    float* ws   = (float*)d_ws;
    float* deg  = ws;                                  // [n]
    float* dinv = deg  + n;                            // [n]
    float* h1   = dinv + n;                            // [n, 256]
    float* agg1 = h1   + (size_t)n * F1;               // [n, 256]
    float* h2   = agg1 + (size_t)n * F1;               // [n, 128]
    float* agg2 = h2   + (size_t)n * F2;               // [n, 128]

    const int TB = 256;
    const int mtiles = (n + 15) / 16;

    // 1) degrees + symmetric normalization
    gcn_deg_init <<<(n + TB - 1) / TB, TB, 0, stream>>>(deg, n);
    gcn_deg_count<<<1024, TB, 0, stream>>>(edst, deg, E);
    gcn_deg_rsqrt<<<(n + TB - 1) / TB, TB, 0, stream>>>(deg, dinv, n);

    // 2) layer 1: h1 = x @ W1 ; agg1 = b1 + dinv^2*h1 ; += edge messages
    gcn_gemm_wmma<<<dim3(mtiles, F1 / 64), 128, 0, stream>>>(
        x, W1, b1, dinv, h1, agg1, n, F_IN, F1, /*relu_in=*/0);
    gcn_scatter<<<2048, TB, 0, stream>>>(esrc, edst, dinv, h1, agg1, E, F1 / 4);

    // 3) layer 2: h2 = relu(agg1) @ W2 ; agg2 = b2 + dinv^2*h2 ; += edge messages
    gcn_gemm_wmma<<<dim3(mtiles, F2 / 64), 128, 0, stream>>>(
        agg1, W2, b2, dinv, h2, agg2, n, F1, F2, /*relu_in=*/1);
    gcn_scatter<<<2048, TB, 0, stream>>>(esrc, edst, dinv, h2, agg2, E, F2 / 4);

    // 4) head: out = relu(agg2) @ Wfc + bfc
    gcn_head<<<1024, TB, 0, stream>>>(agg2, Wfc, bfc, out, n);
}